// CANE_FeatureEmbedding_40037685133334
// MI455X (gfx1250) — compile-verified
//
#include <hip/hip_runtime.h>
#include <math.h>

// ---------------------------------------------------------------------------
// CDNA5 / gfx1250 implementation of CANE FeatureEmbedding.
// Dense GEMM portions use V_WMMA_F32_16X16X4_F32 (exact fp32 matrix core path).
// ---------------------------------------------------------------------------

typedef __attribute__((ext_vector_type(2))) float v2f;
typedef __attribute__((ext_vector_type(8))) float v8f;

__device__ __forceinline__ v8f wmma4(v2f a, v2f b, v8f c) {
  // D = A(16x4,f32) * B(4x16,f32) + C(16x16,f32)
  return __builtin_amdgcn_wmma_f32_16x16x4_f32(
      /*neg_a=*/false, a, /*neg_b=*/false, b,
      /*c_mod=*/(short)0, c, /*reuse_a=*/false, /*reuse_b=*/false);
}

#define OUT_DIM 224

// ---------------------------------------------------------------------------
// 0) zero output + degree buffer
// ---------------------------------------------------------------------------
__global__ __launch_bounds__(256) void k_zero(float* __restrict__ out,
                                              float* __restrict__ deg,
                                              long nOut, int n) {
  long i = (long)blockIdx.x * 256 + threadIdx.x;
  if (i < nOut) out[i] = 0.0f;
  else if (i < nOut + n) deg[i - nOut] = 0.0f;
}

// ---------------------------------------------------------------------------
// 1) degree count per target (row) node
// ---------------------------------------------------------------------------
__global__ __launch_bounds__(256) void k_deg(const int* __restrict__ rowA,
                                             float* __restrict__ deg, int E) {
  int e = blockIdx.x * 256 + threadIdx.x;
  if (e < E) atomicAdd(&deg[rowA[e]], 1.0f);
}

// deg -> deg^-0.5 (0 where deg==0), in place
__global__ __launch_bounds__(256) void k_dinv(float* __restrict__ deg, int n) {
  int i = blockIdx.x * 256 + threadIdx.x;
  if (i < n) {
    float d = deg[i];
    deg[i] = (d > 0.0f) ? (1.0f / sqrtf(d)) : 0.0f;
  }
}

// ---------------------------------------------------------------------------
// 2) ego branch: out[:,0:64] = relu(x @ W_ego^T + b_ego)
//    M=N nodes (16/wave), K=128 (32 wmma steps), Nout=64 (4 tiles)
// ---------------------------------------------------------------------------
#define EGO_PAD 132
__global__ __launch_bounds__(256) void k_ego(const float* __restrict__ x,
                                             const float* __restrict__ W,   // [64,128]
                                             const float* __restrict__ b,   // [64]
                                             float* __restrict__ out,
                                             int numTiles) {
  __shared__ float sW[64 * EGO_PAD];
  for (int i = threadIdx.x; i < 64 * 128; i += 256)
    sW[(i >> 7) * EGO_PAD + (i & 127)] = W[i];
  __syncthreads();

  int wave = threadIdx.x >> 5, lane = threadIdx.x & 31;
  int tile = blockIdx.x * 8 + wave;
  if (tile >= numTiles) return;
  int nodeBase = tile * 16;
  int nlo = lane & 15, half = lane >> 4;

  const float* xb = x + (size_t)(nodeBase + nlo) * 128 + 2 * half;
  const float* wb = &sW[nlo * EGO_PAD + 2 * half];

  const v8f vz = {0.f, 0.f, 0.f, 0.f, 0.f, 0.f, 0.f, 0.f};
  v8f acc[4];
#pragma unroll
  for (int t = 0; t < 4; ++t) acc[t] = vz;

#pragma unroll 4
  for (int s = 0; s < 32; ++s) {
    v2f a = *(const v2f*)(xb + 4 * s);
#pragma unroll
    for (int t = 0; t < 4; ++t) {
      v2f bb = *(const v2f*)(wb + 16 * t * EGO_PAD + 4 * s);
      acc[t] = wmma4(a, bb, acc[t]);
    }
  }

#pragma unroll
  for (int t = 0; t < 4; ++t) {
    int ng = 16 * t + nlo;
    float bias = b[ng];
#pragma unroll
    for (int j = 0; j < 8; ++j) {
      int m = 8 * half + j;
      float v = acc[t][j] + bias;
      out[(size_t)(nodeBase + m) * OUT_DIM + ng] = (v > 0.0f) ? v : 0.0f;
    }
  }
}

// ---------------------------------------------------------------------------
// 3) edge branch: per-edge h_e = relu(edge_attrs @ W_edge^T + b_edge),
//    atomic scatter-sum into out[:,64:96]  (== h_e_agg, no post-activation)
//    M=E edges (16/wave), K=32 (8 steps), Nout=32 (2 tiles)
// ---------------------------------------------------------------------------
#define EDGE_PAD 36
__global__ __launch_bounds__(256) void k_edge(const float* __restrict__ ea,
                                              const float* __restrict__ W,  // [32,32]
                                              const float* __restrict__ b,  // [32]
                                              const int* __restrict__ rowA,
                                              float* __restrict__ out,
                                              int numTiles) {
  __shared__ float sW[32 * EDGE_PAD];
  for (int i = threadIdx.x; i < 32 * 32; i += 256)
    sW[(i >> 5) * EDGE_PAD + (i & 31)] = W[i];
  __syncthreads();

  int wave = threadIdx.x >> 5, lane = threadIdx.x & 31;
  int tile = blockIdx.x * 8 + wave;
  if (tile >= numTiles) return;
  int eBase = tile * 16;
  int nlo = lane & 15, half = lane >> 4;

  const float* eb = ea + (size_t)(eBase + nlo) * 32 + 2 * half;
  const float* wb = &sW[nlo * EDGE_PAD + 2 * half];

  int rj[8];
#pragma unroll
  for (int j = 0; j < 8; ++j) rj[j] = rowA[eBase + 8 * half + j];

  const v8f vz = {0.f, 0.f, 0.f, 0.f, 0.f, 0.f, 0.f, 0.f};
  v8f acc[2];
  acc[0] = vz; acc[1] = vz;

#pragma unroll
  for (int s = 0; s < 8; ++s) {
    v2f a = *(const v2f*)(eb + 4 * s);
#pragma unroll
    for (int t = 0; t < 2; ++t) {
      v2f bb = *(const v2f*)(wb + 16 * t * EDGE_PAD + 4 * s);
      acc[t] = wmma4(a, bb, acc[t]);
    }
  }

#pragma unroll
  for (int t = 0; t < 2; ++t) {
    int ng = 16 * t + nlo;
    float bias = b[ng];
#pragma unroll
    for (int j = 0; j < 8; ++j) {
      float v = acc[t][j] + bias;
      v = (v > 0.0f) ? v : 0.0f;  // relu BEFORE aggregation (per reference)
      atomicAdd(out + (size_t)rj[j] * OUT_DIM + 64 + ng, v);
    }
  }
}

// ---------------------------------------------------------------------------
// 4) peer branch: feat=[x[row] , edge_attrs] (160), msg = norm*(feat@Wp^T+bp),
//    atomic scatter-sum by row into out[:,128:224] (relu applied afterwards)
//    M=E edges (16/wave), K=160 (40 steps: 32 from x, 8 from ea), Nout=96 (6 tiles)
// ---------------------------------------------------------------------------
#define PEER_PAD 164
__global__ __launch_bounds__(256) void k_peer(const float* __restrict__ x,
                                              const float* __restrict__ ea,
                                              const float* __restrict__ Wp,  // [96,160]
                                              const float* __restrict__ bp,  // [96]
                                              const int* __restrict__ rowA,
                                              const float* __restrict__ dinv,
                                              float* __restrict__ out,
                                              int numTiles) {
  __shared__ float sW[96 * PEER_PAD];
  for (int i = threadIdx.x; i < 96 * 160; i += 256)
    sW[(i / 160) * PEER_PAD + (i % 160)] = Wp[i];
  __syncthreads();

  int wave = threadIdx.x >> 5, lane = threadIdx.x & 31;
  int tile = blockIdx.x * 8 + wave;
  if (tile >= numTiles) return;
  int eBase = tile * 16;
  int nlo = lane & 15, half = lane >> 4;

  // A-fragment source pointers for this lane's M-row (m = nlo)
  int eA = eBase + nlo;
  int xrow = rowA[eA];
  const float* xb = x + (size_t)xrow * 128 + 2 * half;
  const float* eb = ea + (size_t)eA * 32 + 2 * half;
  const float* wb = &sW[nlo * PEER_PAD + 2 * half];

  // row index + norm for the M-rows this lane scatters (m = 8*half + j)
  int rj[8];
  float nv[8];
#pragma unroll
  for (int j = 0; j < 8; ++j) {
    int r = rowA[eBase + 8 * half + j];
    rj[j] = r;
    nv[j] = dinv[r];
  }

  const v8f vz = {0.f, 0.f, 0.f, 0.f, 0.f, 0.f, 0.f, 0.f};
  v8f acc[6];
#pragma unroll
  for (int t = 0; t < 6; ++t) acc[t] = vz;

  // K = 0..127 : node features (wave-uniform split, no divergence)
#pragma unroll 4
  for (int s = 0; s < 32; ++s) {
    v2f a = *(const v2f*)(xb + 4 * s);
#pragma unroll
    for (int t = 0; t < 6; ++t) {
      v2f bb = *(const v2f*)(wb + 16 * t * PEER_PAD + 4 * s);
      acc[t] = wmma4(a, bb, acc[t]);
    }
  }
  // K = 128..159 : edge features
#pragma unroll
  for (int s = 0; s < 8; ++s) {
    v2f a = *(const v2f*)(eb + 4 * s);
#pragma unroll
    for (int t = 0; t < 6; ++t) {
      v2f bb = *(const v2f*)(wb + 16 * t * PEER_PAD + 128 + 4 * s);
      acc[t] = wmma4(a, bb, acc[t]);
    }
  }

#pragma unroll
  for (int t = 0; t < 6; ++t) {
    int ng = 16 * t + nlo;
    float bias = bp[ng];
#pragma unroll
    for (int j = 0; j < 8; ++j) {
      float v = (acc[t][j] + bias) * nv[j];  // bias inside norm scaling (per ref)
      atomicAdd(out + (size_t)rj[j] * OUT_DIM + 128 + ng, v);
    }
  }
}

// ---------------------------------------------------------------------------
// 5) h_e_mm: out[row,96:128] += dinv[row] * out[col,64:96]   (after k_edge)
// ---------------------------------------------------------------------------
__global__ __launch_bounds__(256) void k_hemm(const int* __restrict__ colA,
                                              const int* __restrict__ rowA,
                                              const float* __restrict__ dinv,
                                              float* __restrict__ out, int E) {
  long idx = (long)blockIdx.x * 256 + threadIdx.x;
  int e = (int)(idx >> 5);
  int c = (int)(idx & 31);
  if (e < E) {
    int r = rowA[e];
    int s = colA[e];
    float nvv = dinv[r];
    float v = out[(size_t)s * OUT_DIM + 64 + c];
    atomicAdd(out + (size_t)r * OUT_DIM + 96 + c, nvv * v);
  }
}

// ---------------------------------------------------------------------------
// 6) relu on aggregated peer block out[:,128:224]
// ---------------------------------------------------------------------------
__global__ __launch_bounds__(256) void k_relu_peer(float* __restrict__ out, int n) {
  long i = (long)blockIdx.x * 256 + threadIdx.x;
  if (i < (long)n * 96) {
    long node = i / 96;
    int c = (int)(i % 96);
    float* p = out + (size_t)node * OUT_DIM + 128 + c;
    float v = *p;
    *p = (v > 0.0f) ? v : 0.0f;
  }
}

// ---------------------------------------------------------------------------
extern "C" void kernel_launch(void* const* d_in, const int* in_sizes, int n_in,
                              void* d_out, int out_size, void* d_ws, size_t ws_size,
                              hipStream_t stream) {
  (void)n_in; (void)out_size; (void)ws_size;
  const float* x   = (const float*)d_in[0];   // [N,128]
  const float* ea  = (const float*)d_in[1];   // [E,32]
  const float* Wp  = (const float*)d_in[2];   // [96,160]
  const float* bp  = (const float*)d_in[3];   // [96]
  const float* Wg  = (const float*)d_in[4];   // [64,128]
  const float* bg  = (const float*)d_in[5];   // [64]
  const float* Wd  = (const float*)d_in[6];   // [32,32]
  const float* bd  = (const float*)d_in[7];   // [32]
  const int*   ei  = (const int*)d_in[8];     // [2,E] : [0]=col(src), [1]=row(dst)

  const int n = in_sizes[0] / 128;
  const int E = in_sizes[1] / 32;
  const int* colA = ei;
  const int* rowA = ei + E;

  float* out = (float*)d_out;
  float* deg = (float*)d_ws;  // n floats; becomes dinv in place

  const long nOut = (long)n * OUT_DIM;
  const long zTot = nOut + n;

  k_zero<<<(int)((zTot + 255) / 256), 256, 0, stream>>>(out, deg, nOut, n);
  k_deg<<<(E + 255) / 256, 256, 0, stream>>>(rowA, deg, E);
  k_dinv<<<(n + 255) / 256, 256, 0, stream>>>(deg, n);

  const int egoTiles = (n + 15) / 16;     // N=50000 -> 3125 full tiles
  k_ego<<<(egoTiles + 7) / 8, 256, 0, stream>>>(x, Wg, bg, out, egoTiles);

  const int eTiles = (E + 15) / 16;       // E=800000 -> 50000 full tiles
  k_edge<<<(eTiles + 7) / 8, 256, 0, stream>>>(ea, Wd, bd, rowA, out, eTiles);
  k_peer<<<(eTiles + 7) / 8, 256, 0, stream>>>(x, ea, Wp, bp, rowA, deg, out, eTiles);

  const long hemmTot = (long)E * 32;
  k_hemm<<<(int)((hemmTot + 255) / 256), 256, 0, stream>>>(colA, rowA, deg, out, E);

  const long reluTot = (long)n * 96;
  k_relu_peer<<<(int)((reluTot + 255) / 256), 256, 0, stream>>>(out, n);
}